// GNN_7129645711821
// MI455X (gfx1250) — compile-verified
//
#include <hip/hip_runtime.h>
#include <math.h>

// ---------------------------------------------------------------------------
// GAT x3 + BN pipeline for MI455X (gfx1250, wave32).
// Dense x@W uses V_WMMA_F32_16X16X4_F32 (fp32 matrix pipe, exact precision).
// Edge phase: wave-per-edge coalesced gather + f32 atomic scatter (L2-resident).
// ---------------------------------------------------------------------------

typedef float v2f __attribute__((ext_vector_type(2)));
typedef float v8f __attribute__((ext_vector_type(8)));

#define LRELU_SLOPE 0.2f
#define BN_EPS 1e-5f

__device__ __forceinline__ void atomicMaxF32(float* addr, float val) {
    // Sign-aware float atomic max via int/uint ordering trick.
    if (val >= 0.0f) atomicMax((int*)addr, __float_as_int(val));
    else             atomicMin((unsigned int*)addr, __float_as_uint(val));
}

// -------------------- dense GEMM: C[M,Ncols] = A[M,K] * B[K,Ncols] ----------
// One wave computes one 16x16 output tile via V_WMMA_F32_16X16X4_F32.
// A 16x4 layout : lanes 0-15 rows M=lane, v0=K0 v1=K1; lanes 16-31 v0=K2 v1=K3.
// B 4x16 layout : lanes 0-15 cols N=lane, v0=K0 v1=K1; lanes 16-31 v0=K2 v1=K3.
// C/D 16x16     : vgpr i, lanes 0-15 -> M=i, lanes 16-31 -> M=i+8.
// Loads are unconditional (row index clamped, stores guarded) so the hot loop
// carries no EXEC manipulation; K-loop unrolled 8x for load clausing.
__global__ void __launch_bounds__(256)
gemm_wmma_f32(const float* __restrict__ A, const float* __restrict__ B,
              float* __restrict__ C, int M, int K, int Ncols) {
    const int lane = threadIdx.x & 31;
    const int wave = (blockIdx.x * blockDim.x + threadIdx.x) >> 5;
    const int nT   = Ncols >> 4;
    const int mT   = (M + 15) >> 4;
    if (wave >= mT * nT) return;                 // wave-uniform exit (EXEC all-1 inside)
    const int mt   = wave / nT;
    const int m0   = mt << 4;
    const int n0   = (wave - mt * nT) << 4;

    int mrow = m0 + (lane & 15);
    if (mrow > M - 1) mrow = M - 1;              // clamp -> always-legal loads
    const int ncol = n0 + (lane & 15);
    const int koff = (lane >> 4) << 1;           // 0 for lanes 0-15, 2 for 16-31

    const float* Arow = A + (size_t)mrow * K + koff;       // 8B-aligned (K%4==0)
    const float* Bcol = B + (size_t)koff * Ncols + ncol;

    v8f acc = {0.f, 0.f, 0.f, 0.f, 0.f, 0.f, 0.f, 0.f};
#pragma unroll 8
    for (int kb = 0; kb < K; kb += 4) {
        const v2f a = *(const v2f*)(Arow + kb);  // one global_load_b64
        v2f b;
        b.x = Bcol[(size_t)kb * Ncols];
        b.y = Bcol[(size_t)(kb + 1) * Ncols];
        acc = __builtin_amdgcn_wmma_f32_16x16x4_f32(
                  false, a, false, b, (short)0, acc, false, false);
    }

    const int rbase = m0 + ((lane >> 4) << 3);
#pragma unroll
    for (int i = 0; i < 8; ++i) {
        const int r = rbase + i;
        if (r < M) C[(size_t)r * Ncols + ncol] = acc[i];
    }
}

// -------------------- attention logits al_s/al_d [N,H] ----------------------
__global__ void logits_kernel(const float* __restrict__ xw,
                              const float* __restrict__ a_src,
                              const float* __restrict__ a_dst,
                              float* __restrict__ alS, float* __restrict__ alD,
                              int Nn, int H, int C) {
    const int idx = blockIdx.x * blockDim.x + threadIdx.x;
    if (idx >= Nn * H) return;
    const int n = idx / H, h = idx - n * H;
    const float* row = xw + (size_t)n * H * C + h * C;
    float s = 0.f, d = 0.f;
    for (int c = 0; c < C; ++c) {
        const float v = row[c];
        s += v * a_src[h * C + c];
        d += v * a_dst[h * C + c];
    }
    alS[idx] = s;
    alD[idx] = d;
}

// -------------------- per-layer state init ----------------------------------
__global__ void init_kernel(float* __restrict__ gat, const float* __restrict__ bias,
                            float* __restrict__ emax, float* __restrict__ denom,
                            float* __restrict__ stats, int Nn, int H, int HC) {
    const int idx = blockIdx.x * blockDim.x + threadIdx.x;
    const int tot = Nn * HC;
    if (idx < tot)      gat[idx]   = bias[idx % HC];
    if (idx < Nn * H) { emax[idx]  = -INFINITY; denom[idx] = 0.f; }
    if (idx < 2 * HC)   stats[idx] = 0.f;
}

// -------------------- softmax pass 1: segment max ---------------------------
__global__ void edge_max_kernel(const float* __restrict__ alS,
                                const float* __restrict__ alD,
                                const int* __restrict__ adj,
                                float* __restrict__ emax, int E, int Nn, int H) {
    const int idx = blockIdx.x * blockDim.x + threadIdx.x;
    const int tot = (E + Nn) * H;
    if (idx >= tot) return;
    const int e = idx / H, h = idx - e * H;
    int s, d;
    if (e < E) { s = adj[e]; d = adj[E + e]; } else { s = d = e - E; }
    float v = alS[s * H + h] + alD[d * H + h];
    v = v > 0.f ? v : LRELU_SLOPE * v;
    atomicMaxF32(&emax[d * H + h], v);
}

// -------------------- softmax pass 2: segment sum of exp --------------------
__global__ void edge_sum_kernel(const float* __restrict__ alS,
                                const float* __restrict__ alD,
                                const int* __restrict__ adj,
                                const float* __restrict__ emax,
                                float* __restrict__ denom, int E, int Nn, int H) {
    const int idx = blockIdx.x * blockDim.x + threadIdx.x;
    const int tot = (E + Nn) * H;
    if (idx >= tot) return;
    const int e = idx / H, h = idx - e * H;
    int s, d;
    if (e < E) { s = adj[e]; d = adj[E + e]; } else { s = d = e - E; }
    float v = alS[s * H + h] + alD[d * H + h];
    v = v > 0.f ? v : LRELU_SLOPE * v;
    atomicAdd(&denom[d * H + h], expf(v - emax[d * H + h]));
}

// -------------------- weighted message scatter (wave per edge) --------------
__global__ void scatter_kernel(const float* __restrict__ xw,
                               const float* __restrict__ alS,
                               const float* __restrict__ alD,
                               const float* __restrict__ emax,
                               const float* __restrict__ denom,
                               const int* __restrict__ adj,
                               float* __restrict__ gat,
                               int E, int Nn, int H, int C) {
    const int wave = (blockIdx.x * blockDim.x + threadIdx.x) >> 5;
    const int lane = threadIdx.x & 31;
    const int tot  = E + Nn;
    if (wave >= tot) return;                     // wave-uniform
    int s, d;
    if (wave < E) { s = adj[wave]; d = adj[E + wave]; } else { s = d = wave - E; }
    float alpha[2];
    for (int h = 0; h < H; ++h) {
        float v = alS[s * H + h] + alD[d * H + h];
        v = v > 0.f ? v : LRELU_SLOPE * v;
        const float w = expf(v - emax[d * H + h]);
        alpha[h] = w / (denom[d * H + h] + 1e-16f);
    }
    const int HC = H * C;
    const size_t sb = (size_t)s * HC, db = (size_t)d * HC;
    for (int f = lane; f < HC; f += 32)
        atomicAdd(&gat[db + f], xw[sb + f] * alpha[f / C]);
}

// -------------------- BN stats: per-channel sum / sumsq of relu(gat) --------
__global__ void bn_stats_kernel(const float* __restrict__ gat,
                                float* __restrict__ stats, int Nn, int HC) {
    const int c = threadIdx.x;                   // blockDim.x == HC
    float s = 0.f, s2 = 0.f;
    for (int r = blockIdx.x; r < Nn; r += gridDim.x) {
        float v = gat[(size_t)r * HC + c];
        v = v > 0.f ? v : 0.f;
        s += v;
        s2 += v * v;
    }
    atomicAdd(&stats[c], s);
    atomicAdd(&stats[HC + c], s2);
}

// -------------------- BN apply + write next-layer input and out slice -------
__global__ void bn_apply_kernel(const float* __restrict__ gat,
                                const float* __restrict__ stats,
                                const float* __restrict__ g,
                                const float* __restrict__ be,
                                float* __restrict__ xnext, float* __restrict__ out,
                                int Nn, int HC, int outStride, int colOfs) {
    const int idx = blockIdx.x * blockDim.x + threadIdx.x;
    if (idx >= Nn * HC) return;
    const int n = idx / HC, c = idx - n * HC;
    const float invN = 1.f / (float)Nn;
    const float m    = stats[c] * invN;
    const float var  = stats[HC + c] * invN - m * m;
    float v = gat[idx];
    v = v > 0.f ? v : 0.f;
    const float y = (v - m) * rsqrtf(var + BN_EPS) * g[c] + be[c];
    if (xnext) xnext[idx] = y;
    out[(size_t)n * outStride + colOfs + c] = y;
}

// ---------------------------------------------------------------------------
extern "C" void kernel_launch(void* const* d_in, const int* in_sizes, int n_in,
                              void* d_out, int out_size, void* d_ws, size_t ws_size,
                              hipStream_t stream) {
    (void)n_in; (void)out_size; (void)ws_size;
    const float* x   = (const float*)d_in[0];
    const int*   adj = (const int*)  d_in[1];
    const float* W1  = (const float*)d_in[2];
    const float* as1 = (const float*)d_in[3];
    const float* ad1 = (const float*)d_in[4];
    const float* b1  = (const float*)d_in[5];
    const float* g1  = (const float*)d_in[6];
    const float* be1 = (const float*)d_in[7];
    const float* W2  = (const float*)d_in[8];
    const float* as2 = (const float*)d_in[9];
    const float* ad2 = (const float*)d_in[10];
    const float* b2  = (const float*)d_in[11];
    const float* g2  = (const float*)d_in[12];
    const float* be2 = (const float*)d_in[13];
    const float* W3  = (const float*)d_in[14];
    const float* as3 = (const float*)d_in[15];
    const float* ad3 = (const float*)d_in[16];
    const float* b3  = (const float*)d_in[17];
    const float* g3  = (const float*)d_in[18];
    const float* be3 = (const float*)d_in[19];

    const int Nn = in_sizes[0] / 128;   // 50000
    const int E  = in_sizes[1] / 2;     // 800000
    float* out   = (float*)d_out;       // [N, 320]

    // workspace carve-out (all float32), 64-element aligned
    float* base = (float*)d_ws;
    size_t off  = 0;
    auto carve  = [&](size_t nf) { float* p = base + off; off += (nf + 63) & ~(size_t)63; return p; };
    float* bufA  = carve((size_t)Nn * 128);   // layer input / bn output ping
    float* bufB  = carve((size_t)Nn * 128);   // xw
    float* bufC  = carve((size_t)Nn * 128);   // aggregated gat output
    float* alS   = carve((size_t)Nn * 2);
    float* alD   = carve((size_t)Nn * 2);
    float* emax  = carve((size_t)Nn * 2);
    float* denom = carve((size_t)Nn * 2);
    float* stats = carve(256);                // [sum(HC), sumsq(HC)]

    auto run_layer = [&](const float* xin, int Fin,
                         const float* W, const float* as_, const float* ad_,
                         const float* bb, const float* gg, const float* bee,
                         int H, int C, float* xw, float* gat, float* xnext,
                         int colOfs) {
        const int HC  = H * C;
        const int tot = Nn * HC;

        // 1) xw = xin @ W  (fp32 WMMA)
        const int tiles = ((Nn + 15) / 16) * (HC / 16);
        gemm_wmma_f32<<<(tiles + 7) / 8, 256, 0, stream>>>(xin, W, xw, Nn, Fin, HC);

        // 2) attention logits
        logits_kernel<<<(Nn * H + 255) / 256, 256, 0, stream>>>(xw, as_, ad_, alS, alD, Nn, H, C);

        // 3) init gat=bias, emax=-inf, denom=0, stats=0
        init_kernel<<<(tot + 255) / 256, 256, 0, stream>>>(gat, bb, emax, denom, stats, Nn, H, HC);

        // 4-5) segment softmax (max, then sum of exp)
        const int eh = (E + Nn) * H;
        edge_max_kernel<<<(eh + 255) / 256, 256, 0, stream>>>(alS, alD, adj, emax, E, Nn, H);
        edge_sum_kernel<<<(eh + 255) / 256, 256, 0, stream>>>(alS, alD, adj, emax, denom, E, Nn, H);

        // 6) weighted scatter-add, wave per edge
        const int waves = E + Nn;
        scatter_kernel<<<(waves + 7) / 8, 256, 0, stream>>>(xw, alS, alD, emax, denom, adj,
                                                            gat, E, Nn, H, C);

        // 7-8) batchnorm over relu(gat)
        bn_stats_kernel<<<512, HC, 0, stream>>>(gat, stats, Nn, HC);
        bn_apply_kernel<<<(tot + 255) / 256, 256, 0, stream>>>(gat, stats, gg, bee,
                                                               xnext, out, Nn, HC, 320, colOfs);
    };

    // layer 1: 128 -> [H=2,C=64] -> x1 in bufA, out[:,0:128]
    run_layer(x,    128, W1, as1, ad1, b1, g1, be1, 2, 64, bufB, bufC, bufA, 0);
    // layer 2: 128 -> [H=2,C=64] -> x2 back into bufA (free after its GEMM), out[:,128:256]
    run_layer(bufA, 128, W2, as2, ad2, b2, g2, be2, 2, 64, bufB, bufC, bufA, 128);
    // layer 3: 128 -> [H=1,C=64] -> only out[:,256:320]
    run_layer(bufA, 128, W3, as3, ad3, b3, g3, be3, 1, 64, bufB, bufC, nullptr, 256);
}